// RecurrentAdamRAttention_13975823582133
// MI455X (gfx1250) — compile-verified
//
#include <hip/hip_runtime.h>
#include <math.h>

typedef __attribute__((ext_vector_type(2))) float v2f;
typedef __attribute__((ext_vector_type(8))) float v8f;

#define D_DIM 64
#define M_DIM 64

__global__ __launch_bounds__(256) void raatt_kernel(
    const float* __restrict__ q_, const float* __restrict__ k_,
    const float* __restrict__ v_, const float* __restrict__ Si_,
    const float* __restrict__ Zi_, const float* __restrict__ Pi_,
    const float* __restrict__ Mi_, const int* __restrict__ t_,
    float* __restrict__ outV, float* __restrict__ outS,
    float* __restrict__ outZ, float* __restrict__ outP,
    float* __restrict__ outM)
{
    __shared__ __align__(16) float SiF[D_DIM * M_DIM];   // 16 KB: Si_new f32, [d][m]
    __shared__ __align__(16) float Kf[D_DIM];
    __shared__ __align__(16) float Vf[M_DIM];
    __shared__ float Qf[D_DIM];
    __shared__ float zpart[D_DIM];
    __shared__ float Zs;

    const int b   = blockIdx.x;                  // (n,h) pair index
    const int tid = threadIdx.x;
    const size_t base = (size_t)b * (D_DIM * M_DIM);
    const int vb  = b * D_DIM;

    // -- time-dependent uniform scalars (time_indx arrives as device data) --
    const float MU = 0.9f, STEP = 0.1f, BETA = 0.999f;
    const int   t    = t_[0] + 1;
    const float mupt = __powf(MU,   (float)t);
    const float bept = __powf(BETA, (float)t);
    const float c1       = (1.0f - MU) / (STEP * (1.0f - mupt));
    const float bias     = 1.0f - bept;
    const float inv_bias = 1.0f / bias;
    const float rho  = 2.0f / (1.0f - BETA) - 1.0f;
    const float rhoi = rho - 2.0f * (float)t * bept / bias;
    const bool  rect = rhoi > 4.0f;
    float ri = 0.0f;
    if (rect)
        ri = sqrtf((rhoi - 4.0f) * (rhoi - 2.0f) * rho /
                   ((rho - 4.0f) * (rho - 2.0f) * rhoi));

    // -- vector phase: Q, K feature maps, Zi update --
    if (tid < D_DIM) {
        float qv = q_[vb + tid];
        float kv = k_[vb + tid];
        float Qv = qv > 0.0f ? qv + 1.0f : __expf(qv);   // elu(x)+1
        float Kv = kv > 0.0f ? kv + 1.0f : __expf(kv);
        float Zn = Zi_[vb + tid] + Kv;
        Kf[tid] = Kv;
        Vf[tid] = v_[vb + tid];
        Qf[tid] = Qv;
        outZ[vb + tid] = Zn;
        zpart[tid] = Qv * Zn;
    }
    __syncthreads();

    // -- z = Q . Zi_new : wave-0 shuffle reduction --
    if (tid < 32) {
        float s = zpart[tid] + zpart[tid + 32];
        #pragma unroll
        for (int off = 16; off > 0; off >>= 1)
            s += __shfl_down(s, off, 32);
        if (tid == 0) Zs = 1.0f / (s + 1e-6f);
    }

    // -- elementwise streaming phase: 16 elements/thread as 4x float4 --
    #pragma unroll
    for (int it = 0; it < 4; ++it) {
        const int idx = tid * 4 + it * 1024;     // float4-aligned, coalesced
        const int d   = idx >> 6;
        const int m   = idx & 63;
        const float4 sp = *(const float4*)(Si_ + base + idx);
        const float4 pp = *(const float4*)(Pi_ + base + idx);
        const float4 mp = *(const float4*)(Mi_ + base + idx);
        const float  kd = Kf[d];
        const float4 vv = *(const float4*)(Vf + m);

        float4 pn, mn, sn;
        {
            float u;
            u = kd * vv.x; pn.x = (MU*pp.x - STEP*u)*c1; mn.x = BETA*mp.x + (1.0f-BETA)*u*u;
            u = kd * vv.y; pn.y = (MU*pp.y - STEP*u)*c1; mn.y = BETA*mp.y + (1.0f-BETA)*u*u;
            u = kd * vv.z; pn.z = (MU*pp.z - STEP*u)*c1; mn.z = BETA*mp.z + (1.0f-BETA)*u*u;
            u = kd * vv.w; pn.w = (MU*pp.w - STEP*u)*c1; mn.w = BETA*mp.w + (1.0f-BETA)*u*u;
        }
        if (rect) {
            sn.x = sp.x - pn.x * ri * rsqrtf(mn.x * inv_bias + 1e-16f);
            sn.y = sp.y - pn.y * ri * rsqrtf(mn.y * inv_bias + 1e-16f);
            sn.z = sp.z - pn.z * ri * rsqrtf(mn.z * inv_bias + 1e-16f);
            sn.w = sp.w - pn.w * ri * rsqrtf(mn.w * inv_bias + 1e-16f);
        } else {
            sn.x = sp.x - pn.x; sn.y = sp.y - pn.y;
            sn.z = sp.z - pn.z; sn.w = sp.w - pn.w;
        }

        *(float4*)(outP + base + idx) = pn;
        *(float4*)(outM + base + idx) = mn;
        *(float4*)(outS + base + idx) = sn;
        *(float4*)(SiF + idx)         = sn;      // stage for WMMA tail
    }
    __syncthreads();

    // -- WMMA reduction tail (full f32): V[m] = Zs * sum_d Q[d] * Si_new[d][m]
    // waves 0..3 each own one m-block of 16; sixteen accumulating 16x16x4 f32 WMMAs.
    const int wave = tid >> 5;
    const int lane = tid & 31;
    if (wave < 4) {
        const int mb   = wave * 16;
        const int col  = mb + (lane & 15);
        const int koff = (lane < 16) ? 0 : 2;    // A 16x4 / B 4x16 f32 lane layout (ISA 7.12.2)
        v8f acc = {};
        #pragma unroll
        for (int kc = 0; kc < D_DIM; kc += 4) {
            v2f A, B;
            A[0] = Qf[kc + koff + 0];                    // rows broadcast with Q
            A[1] = Qf[kc + koff + 1];
            B[0] = SiF[(kc + koff + 0) * M_DIM + col];   // column 'col'
            B[1] = SiF[(kc + koff + 1) * M_DIM + col];
            acc = __builtin_amdgcn_wmma_f32_16x16x4_f32(
                false, A, false, B, (short)0, acc, false, false);
        }
        // D row 0 lives in VGPR 0 of lanes 0..15 (N = lane)
        if (lane < 16) {
            outV[(size_t)b * M_DIM + mb + lane] = acc[0] * Zs;
        }
    }
}

extern "C" void kernel_launch(void* const* d_in, const int* in_sizes, int n_in,
                              void* d_out, int out_size, void* d_ws, size_t ws_size,
                              hipStream_t stream) {
    const float* q  = (const float*)d_in[0];
    const float* k  = (const float*)d_in[1];
    const float* v  = (const float*)d_in[2];
    const float* Si = (const float*)d_in[3];
    const float* Zi = (const float*)d_in[4];
    const float* Pi = (const float*)d_in[5];
    const float* Mi = (const float*)d_in[6];
    const int*   ti = (const int*)d_in[7];

    const int n_pairs = in_sizes[0] / D_DIM;     // N*H = 4096
    float* out = (float*)d_out;
    // output concat order: V, Si, Zi, Pi, Mi
    size_t oV = 0;
    size_t oS = oV + (size_t)n_pairs * M_DIM;
    size_t oZ = oS + (size_t)n_pairs * D_DIM * M_DIM;
    size_t oP = oZ + (size_t)n_pairs * D_DIM;
    size_t oM = oP + (size_t)n_pairs * D_DIM * M_DIM;

    raatt_kernel<<<n_pairs, 256, 0, stream>>>(
        q, k, v, Si, Zi, Pi, Mi, ti,
        out + oV, out + oS, out + oZ, out + oP, out + oM);
}